// SRBGCN_2954937500463
// MI455X (gfx1250) — compile-verified
//
#include <hip/hip_runtime.h>
#include <math.h>

typedef float v2f __attribute__((ext_vector_type(2)));
typedef float v8f __attribute__((ext_vector_type(8)));

#define N_NODES 16384
#define DIM     32
#define LROWS   16                 // output rows per workgroup
#define KCHUNK  512                // K staged per LDS buffer
#define LSTRIDE (KCHUNK + 4)       // LDS row stride (floats) -> conflict-free
#define NCHUNKS (N_NODES / KCHUNK) // 32
#define NWAVES  8
#define KSUB    (KCHUNK / NWAVES)  // 64 K per wave per chunk
#define KSTEPS  (KSUB / 4)         // 16 WMMA k-steps

// ---------------------------------------------------------------------------
// Fold LW(W) and LWh(h) into T[l] = LW^T @ LWh  (32x32 per layer, row-major)
// ---------------------------------------------------------------------------
__global__ void prep_T(const float* __restrict__ Ws, const float* __restrict__ Hs,
                       float* __restrict__ T) {
  for (int idx = threadIdx.x; idx < 2 * DIM * DIM; idx += blockDim.x) {
    const int l   = idx >> 10;
    const int rem = idx & 1023;
    const int i   = rem >> 5;   // T row (k index of msg contraction)
    const int j   = rem & 31;   // T col
    const float* W = Ws + l * 31 * 31;
    const float* h = Hs + l * DIM;
    const float theta = h[0];
    float ss = 0.f;
    for (int q = 0; q < 31; ++q) { float v = h[1 + q]; ss += v * v; }
    const float invn = 1.0f / sqrtf(ss + 1e-14f);
    const float ch = coshf(theta), sh = sinhf(theta);
    float t;
    if (i == 0) {
      t = (j == 0) ? ch : sh * (h[j] * invn);
    } else {
      float acc = 0.f;
      const float nj = (j == 0) ? 0.f : h[j] * invn;
      for (int k = 1; k < 32; ++k) {
        const float nk = h[k] * invn;
        float lwh;
        if (j == 0) lwh = sh * nk;
        else        lwh = ((k == j) ? 1.0f : 0.0f) - (1.0f - ch) * nk * nj;
        acc += W[(k - 1) * 31 + (i - 1)] * lwh;
      }
      t = acc;
    }
    T[l * 1024 + i * 32 + j] = t;
  }
}

// ---------------------------------------------------------------------------
// msg = x @ T, stored pre-swizzled into the WMMA f32 4x16 B-operand layout.
// One wave per node row; lane = column.
// ---------------------------------------------------------------------------
__global__ void msg_swizzle(const float* __restrict__ x, const float* __restrict__ T,
                            float* __restrict__ msg_swz) {
  const int wave = threadIdx.x >> 5;
  const int lane = threadIdx.x & 31;
  const int r = blockIdx.x * 8 + wave;
  const float* xr = x + (size_t)r * DIM;
  float acc = 0.f;
#pragma unroll
  for (int k = 0; k < DIM; ++k) acc += xr[k] * T[k * DIM + lane];
  const int g = r >> 2, kb = r & 3;
  const int t = lane >> 4, n = lane & 15;
  const int dl = ((kb >> 1) << 4) | n;
  msg_swz[((((g * 2 + t) * 32) + dl) << 1) + (kb & 1)] = acc;
}

// ---------------------------------------------------------------------------
// support = adj @ msg with adj streamed via async direct-to-LDS DMA
// (global_load_async_to_lds_b128, ASYNCcnt), double-buffered, fused epilogue.
// Grid: 1024 WGs x 256 threads (8 wave32). WG owns 16 rows, K split 8-way.
// ---------------------------------------------------------------------------
__global__ void __launch_bounds__(256)
gemm_fused(const float* __restrict__ adj, const float* __restrict__ msg_swz,
           float* __restrict__ xout) {
  __shared__ __align__(16) float lds[2 * LROWS * LSTRIDE];  // ~64.5 KB

  const int tid  = threadIdx.x;
  const int wave = tid >> 5;
  const int lane = tid & 31;
  const int rowBase = blockIdx.x * LROWS;

  // A-operand (16x4 f32) per-lane addressing: row = lane&15, K pair = 2*(lane>>4)
  const int arow = lane & 15;
  const int koff = (lane >> 4) * 2;

  // Staging: thread covers rows {row0 + 2j}, fixed 16B column segment.
  const int row0 = tid >> 7;          // 0..1
  const int col4 = tid & 127;         // 0..127
  const float* gbase0 = adj + (size_t)(rowBase + row0) * N_NODES + col4 * 4;
  const unsigned ldsbyte0 =
      (unsigned)(unsigned long long)&lds[row0 * LSTRIDE + col4 * 4];
  const unsigned ldsbufsz = (unsigned)(LROWS * LSTRIDE * 4);

  v8f acc0 = {};  // N cols 0..15
  v8f acc1 = {};  // N cols 16..31

  // Prologue: DMA chunk 0 into buffer 0
  {
    const float* g = gbase0;
    unsigned l = ldsbyte0;
#pragma unroll
    for (int j = 0; j < 8; ++j) {
      asm volatile("global_load_async_to_lds_b128 %0, %1, off"
                   :: "v"(l), "v"(g) : "memory");
      g += 2 * N_NODES;
      l += 2 * LSTRIDE * 4;
    }
  }
  asm volatile("s_wait_asynccnt 0x0" ::: "memory");
  __syncthreads();

  // Per-wave B pointer: element (c*128 + wave*16 + s, tile t, lane) lives at
  // c*8192 + wave*1024 + s*64 + t*32 + lane  (v2f units)
  const v2f* Bw = (const v2f*)msg_swz + wave * 1024 + lane;

#pragma unroll 1
  for (int c = 0; c < NCHUNKS; ++c) {
    if (c + 1 < NCHUNKS) {           // DMA next chunk into the other buffer
      const float* g = gbase0 + (c + 1) * KCHUNK;
      unsigned l = ldsbyte0 + (((c + 1) & 1) ? ldsbufsz : 0u);
#pragma unroll
      for (int j = 0; j < 8; ++j) {
        asm volatile("global_load_async_to_lds_b128 %0, %1, off"
                     :: "v"(l), "v"(g) : "memory");
        g += 2 * N_NODES;
        l += 2 * LSTRIDE * 4;
      }
    }
    const float* buf = &lds[(c & 1) * LROWS * LSTRIDE];
#pragma unroll
    for (int s = 0; s < KSTEPS; ++s) {
      const v2f a  = *(const v2f*)&buf[arow * LSTRIDE + wave * KSUB + s * 4 + koff];
      const v2f b0 = Bw[s * 64];
      const v2f b1 = Bw[s * 64 + 32];
      acc0 = __builtin_amdgcn_wmma_f32_16x16x4_f32(false, a, false, b0,
                                                   (short)0, acc0, false, false);
      acc1 = __builtin_amdgcn_wmma_f32_16x16x4_f32(false, a, false, b1,
                                                   (short)0, acc1, false, false);
    }
    Bw += 8192;
    asm volatile("s_wait_asynccnt 0x0" ::: "memory");  // our DMA landed
    __syncthreads();                                   // everyone done
  }

  // Cross-wave reduction of 16x32 partials (reuse LDS; 8*512 floats)
  {
    const int half = lane >> 4;          // C/D layout: row = v + 8*(lane>>4)
    const int n    = lane & 15;
#pragma unroll
    for (int v = 0; v < 8; ++v) {
      const int row = v + 8 * half;
      lds[wave * 512 + row * 32 + n]      = acc0[v];
      lds[wave * 512 + row * 32 + 16 + n] = acc1[v];
    }
  }
  __syncthreads();

  // Epilogue: wave w handles rows 2w, 2w+1; lane = column
#pragma unroll
  for (int i = 0; i < 2; ++i) {
    const int row = wave * 2 + i;
    float s = 0.f;
#pragma unroll
    for (int p = 0; p < 8; ++p) s += lds[p * 512 + row * 32 + lane];

    // Minkowski dot: sum_{c>=1} s_c^2 - s_0^2
    float m = (lane == 0) ? -s * s : s * s;
    for (int off = 16; off > 0; off >>= 1) m += __shfl_xor(m, off, 32);
    const float denom = sqrtf(fmaxf(fabsf(m), 1e-8f));
    const float xv = s / denom;
    const float x0 = __shfl(xv, 0, 32);
    const float q  = xv / (x0 + 1.0f);
    float p = 0.f;
    if (lane > 0) {
      const float scale = 1.0507009873554805f, alpha = 1.6732632423543772f;
      p = (q > 0.f) ? scale * q : scale * alpha * (expf(q) - 1.0f);
    }
    float pn = p * p;
    for (int off = 16; off > 0; off >>= 1) pn += __shfl_xor(pn, off, 32);
    const float invd = 1.0f / (1.0f - pn);
    const float o = (lane == 0) ? (1.0f + pn) * invd : 2.0f * p * invd;
    xout[(size_t)(rowBase + row) * DIM + lane] = o;
  }
}

// ---------------------------------------------------------------------------
extern "C" void kernel_launch(void* const* d_in, const int* in_sizes, int n_in,
                              void* d_out, int out_size, void* d_ws, size_t ws_size,
                              hipStream_t stream) {
  const float* node = (const float*)d_in[0];   // (16384, 32)
  const float* adj  = (const float*)d_in[1];   // (16384, 16384)
  const float* Ws   = (const float*)d_in[2];   // (2, 31, 31)
  const float* Hs   = (const float*)d_in[3];   // (2, 32)
  float* out = (float*)d_out;
  float* ws  = (float*)d_ws;

  float* T   = ws;                          // 2 * 1024 floats
  float* msg = ws + 2048;                   // 16384*32 floats (swizzled B)
  float* x1  = msg + (size_t)N_NODES * DIM; // 16384*32 floats

  prep_T<<<1, 256, 0, stream>>>(Ws, Hs, T);

  // Layer 0
  msg_swizzle<<<N_NODES / 8, 256, 0, stream>>>(node, T, msg);
  gemm_fused <<<N_NODES / LROWS, 256, 0, stream>>>(adj, msg, x1);

  // Layer 1 -> final output
  msg_swizzle<<<N_NODES / 8, 256, 0, stream>>>(x1, T + 1024, msg);
  gemm_fused <<<N_NODES / LROWS, 256, 0, stream>>>(adj, msg, out);
}